// Seq2seqPredictor_70385924047214
// MI455X (gfx1250) — compile-verified
//
#include <hip/hip_runtime.h>
#include <hip/hip_bf16.h>
#include <math.h>

typedef float v2f __attribute__((ext_vector_type(2)));
typedef float v8f __attribute__((ext_vector_type(8)));

namespace {
constexpr int cB = 64, cL = 32, cD = 512, cV = 10000, cK = 8;
constexpr int cBK = cB * cK;            // 512
constexpr int cMEM = 4;
constexpr int cLM1 = cL - 1;            // 31
constexpr int TPAD = 0, TEOS = 2, TBOS = 1;
constexpr float NEGV = -1e30f;
constexpr float INV_SQRT_D = 0.044194173824159216f; // 1/sqrt(512)
}

// ---------------------------------------------------------------------------
// WMMA f32 16x16x4 multi-tile helpers: one wave computes NT adjacent 16x16
// C tiles, sharing the A fragment and keeping NT independent accumulation
// chains in flight (hides XDL accumulate latency).
// A layout: lane<16 row M=l16, VGPR0=K(2*half), VGPR1=K(2*half+1)
// C/D layout: element j of v8f is row (j + 8*half), col l16.
// ---------------------------------------------------------------------------
template <int NT, typename FA, typename FB>
__device__ __forceinline__ void wmma_mt(int Ktot, FA fa, FB fb, v8f* acc) {
  const int lane = threadIdx.x & 31;
  const int half = lane >> 4;
  const int l16 = lane & 15;
  for (int k0 = 0; k0 < Ktot; k0 += 4) {
    const int ka = k0 + 2 * half;
    v2f a;
    a[0] = fa(l16, ka);
    a[1] = fa(l16, ka + 1);
#pragma unroll
    for (int t = 0; t < NT; ++t) {
      v2f b;
      b[0] = fb(ka, t, l16);
      b[1] = fb(ka + 1, t, l16);
      acc[t] = __builtin_amdgcn_wmma_f32_16x16x4_f32(false, a, false, b,
                                                     (short)0, acc[t], false,
                                                     false);
    }
  }
}

template <int NT, typename FA0, typename FB0, typename FA1, typename FB1>
__device__ __forceinline__ void wmma_mt2(int Ktot, FA0 fa0, FB0 fb0, FA1 fa1,
                                         FB1 fb1, v8f* acc) {
  const int lane = threadIdx.x & 31;
  const int half = lane >> 4;
  const int l16 = lane & 15;
  for (int k0 = 0; k0 < Ktot; k0 += 4) {
    const int ka = k0 + 2 * half;
    v2f a;
    a[0] = fa0(l16, ka);
    a[1] = fa0(l16, ka + 1);
#pragma unroll
    for (int t = 0; t < NT; ++t) {
      v2f b;
      b[0] = fb0(ka, t, l16);
      b[1] = fb0(ka + 1, t, l16);
      acc[t] = __builtin_amdgcn_wmma_f32_16x16x4_f32(false, a, false, b,
                                                     (short)0, acc[t], false,
                                                     false);
    }
    a[0] = fa1(l16, ka);
    a[1] = fa1(l16, ka + 1);
#pragma unroll
    for (int t = 0; t < NT; ++t) {
      v2f b;
      b[0] = fb1(ka, t, l16);
      b[1] = fb1(ka + 1, t, l16);
      acc[t] = __builtin_amdgcn_wmma_f32_16x16x4_f32(false, a, false, b,
                                                     (short)0, acc[t], false,
                                                     false);
    }
  }
}

// ---------------------------------------------------------------------------
// Encoder: context = tanh(Emb[source] @ We).  (B*L=2048) x 512 x 512
// One wave -> 16 rows x 64 cols (4 tiles).
// ---------------------------------------------------------------------------
__global__ void k_context(const int* __restrict__ src,
                          const float* __restrict__ Emb,
                          const float* __restrict__ We,
                          float* __restrict__ ctx) {
  const int job = blockIdx.x * 4 + (threadIdx.x >> 5);  // 0..1023
  const int tm = job >> 3;   // 0..127
  const int ng = job & 7;    // 0..7 (64-col group)
  const int rowBase = tm * 16, colBase = ng * 64;
  v8f acc[4] = {};
  wmma_mt<4>(
      cD,
      [&](int r, int k) { return Emb[(size_t)src[rowBase + r] * cD + k]; },
      [&](int k, int t, int n) { return We[k * cD + colBase + t * 16 + n]; },
      acc);
  const int lane = threadIdx.x & 31, half = lane >> 4, l16 = lane & 15;
#pragma unroll
  for (int t = 0; t < 4; ++t)
#pragma unroll
    for (int j = 0; j < 8; ++j) {
      const int m = j + 8 * half;
      ctx[(size_t)(rowBase + m) * cD + colBase + t * 16 + l16] =
          tanhf(acc[t][j]);
    }
}

__global__ void k_meanctx(const float* __restrict__ ctx,
                          float* __restrict__ meanc) {
  const int idx = blockIdx.x * 256 + threadIdx.x;  // b*cD + d
  const int b = idx / cD, d = idx % cD;
  float s = 0.f;
  for (int l = 0; l < cL; ++l) s += ctx[((size_t)b * cL + l) * cD + d];
  meanc[idx] = s * (1.0f / cL);
}

// state0 = tanh(mean @ Ws), replicated K times into h.  64 x 512 x 512
__global__ void k_state0(const float* __restrict__ meanc,
                         const float* __restrict__ Ws,
                         float* __restrict__ h) {
  const int job = blockIdx.x * 4 + (threadIdx.x >> 5);  // 0..31
  const int tm = job >> 3, ng = job & 7;
  const int colBase = ng * 64;
  v8f acc[4] = {};
  wmma_mt<4>(
      cD,
      [&](int r, int k) { return meanc[(size_t)(tm * 16 + r) * cD + k]; },
      [&](int k, int t, int n) { return Ws[k * cD + colBase + t * 16 + n]; },
      acc);
  const int lane = threadIdx.x & 31, half = lane >> 4, l16 = lane & 15;
#pragma unroll
  for (int t = 0; t < 4; ++t)
#pragma unroll
    for (int j = 0; j < 8; ++j) {
      const int brow = tm * 16 + j + 8 * half;
      const float v = tanhf(acc[t][j]);
      for (int kk = 0; kk < cK; ++kk)
        h[((size_t)brow * cK + kk) * cD + colBase + t * 16 + l16] = v;
    }
}

// ---------------------------------------------------------------------------
// Static masks
// ---------------------------------------------------------------------------
__global__ void k_init_pat(const int* __restrict__ src,
                           unsigned char* __restrict__ pat) {
  const int idx = blockIdx.x * 256 + threadIdx.x;
  if (idx >= cB * cLM1 * cLM1) return;
  const int b = idx / (cLM1 * cLM1);
  const int rem = idx % (cLM1 * cLM1);
  const int i = rem / cLM1, j = rem % cLM1;
  const int ti = src[b * cL + i + 1], tj = src[b * cL + j + 1];
  pat[idx] = (ti != TPAD && tj != TPAD && ti == tj) ? 1 : 0;
}

__global__ void k_init_known(const unsigned char* __restrict__ pat,
                             const int* __restrict__ src_len,
                             unsigned char* __restrict__ known) {
  const int idx = blockIdx.x * 256 + threadIdx.x;
  if (idx >= cB * cLM1) return;
  const int b = idx / cLM1, i = idx % cLM1;
  int kn = (i == src_len[b] - 2) ? 1 : 0;
  for (int j = 0; j < i; ++j) kn |= pat[b * cLM1 * cLM1 + i * cLM1 + j];
  known[idx] = (unsigned char)kn;
}

__global__ void k_init_state(const int* __restrict__ src_len,
                             float* __restrict__ scores,
                             int* __restrict__ out,
                             int* __restrict__ inp,
                             unsigned char* __restrict__ ban,
                             float* __restrict__ m0) {
  const int idx = blockIdx.x * 256 + threadIdx.x;  // covers cBK*cV exactly
  ban[idx] = ((idx % cV) < 4) ? 1 : 0;             // SPECIAL = {0,1,2,3}
  if (idx < cBK * cMEM * cD) m0[idx] = 0.f;
  if (idx < cB * cK * cLM1) {
    const int b = idx / (cK * cLM1);
    const int j = idx % cLM1;
    out[idx] = (j == src_len[b] - 2) ? TEOS : TPAD;
  }
  if (idx < cBK) {
    inp[idx] = TBOS;
    scores[idx] = ((idx % cK) == 0) ? 0.f : NEGV;
  }
}

// ---------------------------------------------------------------------------
// Decode step kernels
// ---------------------------------------------------------------------------
__global__ void k_embed(const int* __restrict__ inp,
                        const float* __restrict__ Emb,
                        float* __restrict__ e) {
  const int idx = blockIdx.x * 256 + threadIdx.x;
  const int bk = idx / cD, d = idx % cD;
  e[idx] = Emb[(size_t)inp[bk] * cD + d];
}

// z = sigmoid(e@Wz + h@Uz);  r = sigmoid(e@Wr + h@Ur)
__global__ void k_gru_zr(const float* __restrict__ e, const float* __restrict__ h,
                         const float* __restrict__ Wz, const float* __restrict__ Uz,
                         const float* __restrict__ Wr, const float* __restrict__ Ur,
                         float* __restrict__ z, float* __restrict__ r) {
  const int job = blockIdx.x * 4 + (threadIdx.x >> 5);  // 0..511
  const int which = job >> 8;                           // 0:z 1:r
  const int t0 = job & 255;
  const int tm = t0 >> 3, ng = t0 & 7;
  const int colBase = ng * 64;
  const float* W = which ? Wr : Wz;
  const float* U = which ? Ur : Uz;
  float* outp = which ? r : z;
  v8f acc[4] = {};
  wmma_mt2<4>(
      cD,
      [&](int rr, int k) { return e[(size_t)(tm * 16 + rr) * cD + k]; },
      [&](int k, int t, int n) { return W[k * cD + colBase + t * 16 + n]; },
      [&](int rr, int k) { return h[(size_t)(tm * 16 + rr) * cD + k]; },
      [&](int k, int t, int n) { return U[k * cD + colBase + t * 16 + n]; },
      acc);
  const int lane = threadIdx.x & 31, half = lane >> 4, l16 = lane & 15;
#pragma unroll
  for (int t = 0; t < 4; ++t)
#pragma unroll
    for (int j = 0; j < 8; ++j) {
      const int row = tm * 16 + j + 8 * half;
      const int col = colBase + t * 16 + l16;
      outp[(size_t)row * cD + col] = 1.0f / (1.0f + expf(-acc[t][j]));
    }
}

// n = tanh(e@Wn + (r*h)@Un);  hn = (1-z)*n + z*h
__global__ void k_gru_hn(const float* __restrict__ e, const float* __restrict__ h,
                         const float* __restrict__ r, const float* __restrict__ z,
                         const float* __restrict__ Wn, const float* __restrict__ Un,
                         float* __restrict__ hn) {
  const int job = blockIdx.x * 4 + (threadIdx.x >> 5);  // 0..255
  const int tm = job >> 3, ng = job & 7;
  const int colBase = ng * 64;
  v8f acc[4] = {};
  wmma_mt2<4>(
      cD,
      [&](int rr, int k) { return e[(size_t)(tm * 16 + rr) * cD + k]; },
      [&](int k, int t, int n) { return Wn[k * cD + colBase + t * 16 + n]; },
      [&](int rr, int k) {
        const size_t o = (size_t)(tm * 16 + rr) * cD + k;
        return r[o] * h[o];
      },
      [&](int k, int t, int n) { return Un[k * cD + colBase + t * 16 + n]; },
      acc);
  const int lane = threadIdx.x & 31, half = lane >> 4, l16 = lane & 15;
#pragma unroll
  for (int t = 0; t < 4; ++t)
#pragma unroll
    for (int j = 0; j < 8; ++j) {
      const int row = tm * 16 + j + 8 * half;
      const int col = colBase + t * 16 + l16;
      const float nval = tanhf(acc[t][j]);
      const float zz = z[(size_t)row * cD + col];
      const float hh = h[(size_t)row * cD + col];
      hn[(size_t)row * cD + col] = (1.0f - zz) * nval + zz * hh;
    }
}

// attention + context vector (tiny; VALU)
__global__ void k_attn(const float* __restrict__ hn,
                       const float* __restrict__ ctx,
                       const int* __restrict__ src,
                       float* __restrict__ ctxv) {
  __shared__ float hns[cD];
  __shared__ float att[cL];
  __shared__ float w[cL];
  __shared__ float invsum;
  const int bk = blockIdx.x;
  const int b = bk / cK;
  const int tid = threadIdx.x;
  for (int d = tid; d < cD; d += 256) hns[d] = hn[(size_t)bk * cD + d];
  __syncthreads();
  const int s = tid >> 3, sub = tid & 7;
  float p = 0.f;
  for (int d = sub; d < cD; d += 8)
    p += hns[d] * ctx[((size_t)b * cL + s) * cD + d];
  p += __shfl_xor(p, 1, 8);
  p += __shfl_xor(p, 2, 8);
  p += __shfl_xor(p, 4, 8);
  if (sub == 0) {
    float a = p * INV_SQRT_D;
    if (src[b * cL + s] == TPAD) a = NEGV;
    att[s] = a;
  }
  __syncthreads();
  if (tid == 0) {
    float mx = att[0];
    for (int ss = 1; ss < cL; ++ss) mx = fmaxf(mx, att[ss]);
    float sum = 0.f;
    for (int ss = 0; ss < cL; ++ss) {
      const float ev = expf(att[ss] - mx);
      w[ss] = ev;
      sum += ev;
    }
    invsum = 1.0f / sum;
  }
  __syncthreads();
  const float inv = invsum;
  for (int d = tid; d < cD; d += 256) {
    float a = 0.f;
    for (int ss = 0; ss < cL; ++ss)
      a += w[ss] * inv * ctx[((size_t)b * cL + ss) * cD + d];
    ctxv[(size_t)bk * cD + d] = a;
  }
}

// fh = tanh([hn|ctx] @ Wc)   512 x 1024 x 512
__global__ void k_fc(const float* __restrict__ hn, const float* __restrict__ ctxv,
                     const float* __restrict__ Wc, float* __restrict__ fh) {
  const int job = blockIdx.x * 4 + (threadIdx.x >> 5);  // 0..255
  const int tm = job >> 3, ng = job & 7;
  const int colBase = ng * 64;
  v8f acc[4] = {};
  wmma_mt<4>(
      2 * cD,
      [&](int rr, int k) {
        const size_t row = tm * 16 + rr;
        return (k < cD) ? hn[row * cD + k] : ctxv[row * cD + (k - cD)];
      },
      [&](int k, int t, int n) { return Wc[k * cD + colBase + t * 16 + n]; },
      acc);
  const int lane = threadIdx.x & 31, half = lane >> 4, l16 = lane & 15;
#pragma unroll
  for (int t = 0; t < 4; ++t)
#pragma unroll
    for (int j = 0; j < 8; ++j) {
      const int row = tm * 16 + j + 8 * half;
      const int col = colBase + t * 16 + l16;
      fh[(size_t)row * cD + col] = tanhf(acc[t][j]);
    }
}

// shift memory and compute mean:  m0B = [m0A[1:], e];  mm = mean(m0B)
__global__ void k_mem(const float* __restrict__ m0A, const float* __restrict__ e,
                      float* __restrict__ m0B, float* __restrict__ mm) {
  const int idx = blockIdx.x * 256 + threadIdx.x;
  const int bk = idx / cD, d = idx % cD;
  const size_t base = (size_t)bk * cMEM * cD + d;
  const float a1 = m0A[base + 1 * cD];
  const float a2 = m0A[base + 2 * cD];
  const float a3 = m0A[base + 3 * cD];
  const float ev = e[idx];
  m0B[base + 0 * cD] = a1;
  m0B[base + 1 * cD] = a2;
  m0B[base + 2 * cD] = a3;
  m0B[base + 3 * cD] = ev;
  mm[idx] = (a1 + a2 + a3 + ev) * 0.25f;
}

// feat = fh + tanh(mm @ Wm)
__global__ void k_feat(const float* __restrict__ mm, const float* __restrict__ Wm,
                       const float* __restrict__ fh, float* __restrict__ feat) {
  const int job = blockIdx.x * 4 + (threadIdx.x >> 5);  // 0..255
  const int tm = job >> 3, ng = job & 7;
  const int colBase = ng * 64;
  v8f acc[4] = {};
  wmma_mt<4>(
      cD,
      [&](int rr, int k) { return mm[(size_t)(tm * 16 + rr) * cD + k]; },
      [&](int k, int t, int n) { return Wm[k * cD + colBase + t * 16 + n]; },
      acc);
  const int lane = threadIdx.x & 31, half = lane >> 4, l16 = lane & 15;
#pragma unroll
  for (int t = 0; t < 4; ++t)
#pragma unroll
    for (int j = 0; j < 8; ++j) {
      const int row = tm * 16 + j + 8 * half;
      const int col = colBase + t * 16 + l16;
      feat[(size_t)row * cD + col] =
          fh[(size_t)row * cD + col] + tanhf(acc[t][j]);
    }
}

// ---------------------------------------------------------------------------
// logits = feat @ Wo   512 x 512 x 10000  (dominant GEMM; Wo is L2 resident)
// 5 waves / block share one 16-row A stripe (contiguous 32KB of feat) staged
// into LDS via the CDNA5 async-to-LDS DMA path (ASYNCcnt).  Each wave then
// computes a 16x80 output strip: 5 independent WMMA accumulation chains fed
// by ds_load for A and streaming global loads of Wo for B.
// ---------------------------------------------------------------------------
__global__ __launch_bounds__(160) void k_logits(const float* __restrict__ feat,
                                                const float* __restrict__ Wo,
                                                float* __restrict__ lg) {
  __shared__ float As[16 * cD];  // 32 KB
  const int tm = blockIdx.x / 25;   // 0..31
  const int g25 = blockIdx.x % 25;  // 0..24
  const int wave = threadIdx.x >> 5;
  const int lane = threadIdx.x & 31;
  const int half = lane >> 4, l16 = lane & 15;
  // Stage A stripe (feat rows tm*16 .. tm*16+15 are one contiguous block).
  {
    const unsigned asBase = (unsigned)(uintptr_t)(void*)As;  // LDS byte offset
    const float* gbase = feat + (size_t)tm * 16 * cD;
    for (int idx = threadIdx.x; idx < (16 * cD) / 4; idx += 160) {
      const unsigned lds = asBase + idx * 16;
      const float* g = gbase + idx * 4;
      asm volatile("global_load_async_to_lds_b128 %0, %1, off" ::"v"(lds),
                   "v"(g)
                   : "memory");
    }
    asm volatile("s_wait_asynccnt 0x0" ::: "memory");
  }
  __syncthreads();
  const int ng = g25 * 5 + wave;  // 0..124
  const int colBase = ng * 80;    // 5 tiles of 16 columns
  v8f acc[5] = {};
  for (int k0 = 0; k0 < cD; k0 += 4) {
    const int ka = k0 + 2 * half;
    v2f a;
    a[0] = As[l16 * cD + ka];
    a[1] = As[l16 * cD + ka + 1];
#pragma unroll
    for (int t = 0; t < 5; ++t) {
      v2f b;
      b[0] = Wo[(size_t)ka * cV + colBase + t * 16 + l16];
      b[1] = Wo[(size_t)(ka + 1) * cV + colBase + t * 16 + l16];
      acc[t] = __builtin_amdgcn_wmma_f32_16x16x4_f32(false, a, false, b,
                                                     (short)0, acc[t], false,
                                                     false);
    }
  }
#pragma unroll
  for (int t = 0; t < 5; ++t)
#pragma unroll
    for (int j = 0; j < 8; ++j) {
      const int row = tm * 16 + j + 8 * half;
      const int col = colBase + t * 16 + l16;
      lg[(size_t)row * cV + col] = acc[t][j];
    }
}

__global__ void k_logsoftmax(float* __restrict__ lg) {
  __shared__ float red[256];
  const int row = blockIdx.x, tid = threadIdx.x;
  float mx = -INFINITY;
  for (int v = tid; v < cV; v += 256)
    mx = fmaxf(mx, lg[(size_t)row * cV + v]);
  red[tid] = mx;
  __syncthreads();
  for (int off = 128; off >= 1; off >>= 1) {
    if (tid < off) red[tid] = fmaxf(red[tid], red[tid + off]);
    __syncthreads();
  }
  mx = red[0];
  __syncthreads();
  float s = 0.f;
  for (int v = tid; v < cV; v += 256) s += expf(lg[(size_t)row * cV + v] - mx);
  red[tid] = s;
  __syncthreads();
  for (int off = 128; off >= 1; off >>= 1) {
    if (tid < off) red[tid] += red[tid + off];
    __syncthreads();
  }
  const float lse = mx + logf(red[0]);
  for (int v = tid; v < cV; v += 256) lg[(size_t)row * cV + v] -= lse;
}

// ---------------------------------------------------------------------------
// top-8 over K*V = 80000 per batch.  Ties -> lower flat index (jax top_k).
// ---------------------------------------------------------------------------
__device__ __forceinline__ bool tk_better(float v1, int i1, float v2, int i2) {
  return (v1 > v2) || (v1 == v2 && i1 < i2);
}

__global__ void k_topk(const float* __restrict__ logp, float* __restrict__ scores,
                       const int* __restrict__ outCur,
                       const unsigned char* __restrict__ known,
                       const unsigned char* __restrict__ ban,
                       const int* __restrict__ src, int step,
                       int* __restrict__ kidx, int* __restrict__ sym) {
  const int b = blockIdx.x, tid = threadIdx.x;
  const bool ended = (src[b * cL + step + 1] == TPAD);
  if (ended) {
    if (tid < cK) {
      kidx[b * cK + tid] = tid;
      sym[b * cK + tid] = TPAD;
    }
    return;
  }
  __shared__ float lv[256 * 8];
  __shared__ int li[256 * 8];
  const int kn = known[b * cLM1 + step];
  float sc[cK];
  int ct[cK];
#pragma unroll
  for (int k = 0; k < cK; ++k) {
    sc[k] = scores[b * cK + k];
    ct[k] = outCur[(b * cK + k) * cLM1 + step];
  }
  float tv[8];
  int ti[8];
#pragma unroll
  for (int j = 0; j < 8; ++j) {
    tv[j] = -INFINITY;
    ti[j] = 0x7FFFFFFF;
  }
  for (int idx = tid; idx < cK * cV; idx += 256) {
    const int k = idx / cV, v = idx - k * cV;
    const bool allowed =
        kn ? (v == ct[k]) : (ban[(size_t)(b * cK + k) * cV + v] == 0);
    const float val =
        allowed ? (sc[k] + logp[(size_t)(b * cK + k) * cV + v]) : NEGV;
    if (tk_better(val, idx, tv[7], ti[7])) {
      tv[7] = val;
      ti[7] = idx;
#pragma unroll
      for (int j = 7; j > 0; --j) {
        if (tk_better(tv[j], ti[j], tv[j - 1], ti[j - 1])) {
          const float fv = tv[j];
          tv[j] = tv[j - 1];
          tv[j - 1] = fv;
          const int iv = ti[j];
          ti[j] = ti[j - 1];
          ti[j - 1] = iv;
        } else
          break;
      }
    }
  }
#pragma unroll
  for (int j = 0; j < 8; ++j) {
    lv[tid * 8 + j] = tv[j];
    li[tid * 8 + j] = ti[j];
  }
  __syncthreads();
  for (int off = 128; off >= 1; off >>= 1) {
    if (tid < off) {
      float av[8], bv[8], ov[8];
      int ai[8], bi[8], oi[8];
#pragma unroll
      for (int j = 0; j < 8; ++j) {
        av[j] = lv[tid * 8 + j];
        ai[j] = li[tid * 8 + j];
        bv[j] = lv[(tid + off) * 8 + j];
        bi[j] = li[(tid + off) * 8 + j];
      }
      int pa = 0, pb = 0;
#pragma unroll
      for (int j = 0; j < 8; ++j) {
        if (tk_better(av[pa], ai[pa], bv[pb], bi[pb])) {
          ov[j] = av[pa];
          oi[j] = ai[pa];
          ++pa;
        } else {
          ov[j] = bv[pb];
          oi[j] = bi[pb];
          ++pb;
        }
      }
#pragma unroll
      for (int j = 0; j < 8; ++j) {
        lv[tid * 8 + j] = ov[j];
        li[tid * 8 + j] = oi[j];
      }
    }
    __syncthreads();
  }
  if (tid < cK) {
    const float val = lv[tid];
    const int idx = li[tid];
    const int ks = idx / cV, vs = idx - ks * cV;
    kidx[b * cK + tid] = ks;
    sym[b * cK + tid] = vs;
    scores[b * cK + tid] = val;
  }
}

// ---------------------------------------------------------------------------
// Beam-permutation gathers
// ---------------------------------------------------------------------------
__global__ void k_gather_state(const float* __restrict__ hn,
                               const float* __restrict__ m0B,
                               const int* __restrict__ kidx,
                               float* __restrict__ h,
                               float* __restrict__ m0A) {
  const int idx = blockIdx.x * 256 + threadIdx.x;  // cBK*cMEM*cD exact
  const int bk = idx / (cMEM * cD);
  const int rest = idx % (cMEM * cD);
  const int b = bk / cK, kk = bk % cK;
  const int comb = b * cK + kidx[b * cK + kk];
  m0A[idx] = m0B[(size_t)comb * cMEM * cD + rest];
  if (rest < cD) h[(size_t)bk * cD + rest] = hn[(size_t)comb * cD + rest];
}

__global__ void k_gather_ban(const unsigned char* __restrict__ banSrc,
                             const int* __restrict__ kidx,
                             const int* __restrict__ sym,
                             unsigned char* __restrict__ banDst,
                             int* __restrict__ inp) {
  const int idx = blockIdx.x * 256 + threadIdx.x;  // cBK*cV exact
  const int bk = idx / cV, v = idx % cV;
  const int b = bk / cK, kk = bk % cK;
  const int comb = b * cK + kidx[b * cK + kk];
  const int s = sym[b * cK + kk];
  unsigned char val = banSrc[(size_t)comb * cV + v];
  if (v == s) val = 1;
  banDst[idx] = val;
  if (v == 0) inp[bk] = s;
}

__global__ void k_gather_out(const int* __restrict__ outSrc,
                             const int* __restrict__ kidx,
                             const int* __restrict__ sym,
                             const unsigned char* __restrict__ pat, int step,
                             int* __restrict__ outDst) {
  const int idx = blockIdx.x * 256 + threadIdx.x;
  if (idx >= cB * cK * cLM1) return;
  const int b = idx / (cK * cLM1);
  const int rem = idx % (cK * cLM1);
  const int kk = rem / cLM1, j = rem % cLM1;
  const int ks = kidx[b * cK + kk];
  int val = outSrc[(b * cK + ks) * cLM1 + j];
  if (pat[b * cLM1 * cLM1 + step * cLM1 + j]) val = sym[b * cK + kk];
  outDst[idx] = val;
}

__global__ void k_finalize(const int* __restrict__ outF,
                           const float* __restrict__ scores,
                           float* __restrict__ dout) {
  const int idx = blockIdx.x * 256 + threadIdx.x;
  if (idx < cB * cK * cLM1)
    dout[idx] = (float)outF[idx];
  else if (idx < cB * cK * cLM1 + cB * cK)
    dout[idx] = scores[idx - cB * cK * cLM1];
}

// ---------------------------------------------------------------------------
extern "C" void kernel_launch(void* const* d_in, const int* in_sizes, int n_in,
                              void* d_out, int out_size, void* d_ws,
                              size_t ws_size, hipStream_t stream) {
  (void)in_sizes;
  (void)n_in;
  (void)out_size;
  (void)ws_size;
  const int* src = (const int*)d_in[0];
  const int* src_len = (const int*)d_in[1];
  const float* Emb = (const float*)d_in[3];
  const float* We = (const float*)d_in[4];
  const float* Ws = (const float*)d_in[5];
  const float* Wz = (const float*)d_in[6];
  const float* Uz = (const float*)d_in[7];
  const float* Wr = (const float*)d_in[8];
  const float* Ur = (const float*)d_in[9];
  const float* Wn = (const float*)d_in[10];
  const float* Un = (const float*)d_in[11];
  const float* Wc = (const float*)d_in[12];
  const float* Wm = (const float*)d_in[13];
  const float* Wo = (const float*)d_in[14];

  char* p = (char*)d_ws;
  auto alloc = [&](size_t bytes) -> void* {
    void* r = (void*)p;
    p += (bytes + 255) & ~(size_t)255;
    return r;
  };
  float* context = (float*)alloc(sizeof(float) * (size_t)cB * cL * cD);
  float* meanc = (float*)alloc(sizeof(float) * cB * cD);
  float* h = (float*)alloc(sizeof(float) * (size_t)cBK * cD);
  float* hn = (float*)alloc(sizeof(float) * (size_t)cBK * cD);
  float* e = (float*)alloc(sizeof(float) * (size_t)cBK * cD);
  float* z = (float*)alloc(sizeof(float) * (size_t)cBK * cD);
  float* r = (float*)alloc(sizeof(float) * (size_t)cBK * cD);
  float* ctxv = (float*)alloc(sizeof(float) * (size_t)cBK * cD);
  float* fh = (float*)alloc(sizeof(float) * (size_t)cBK * cD);
  float* mm = (float*)alloc(sizeof(float) * (size_t)cBK * cD);
  float* feat = (float*)alloc(sizeof(float) * (size_t)cBK * cD);
  float* m0A = (float*)alloc(sizeof(float) * (size_t)cBK * cMEM * cD);
  float* m0B = (float*)alloc(sizeof(float) * (size_t)cBK * cMEM * cD);
  float* logits = (float*)alloc(sizeof(float) * (size_t)cBK * cV);
  float* scores = (float*)alloc(sizeof(float) * cBK);
  int* out0 = (int*)alloc(sizeof(int) * cB * cK * cLM1);
  int* out1 = (int*)alloc(sizeof(int) * cB * cK * cLM1);
  int* inp = (int*)alloc(sizeof(int) * cBK);
  int* kidx = (int*)alloc(sizeof(int) * cB * cK);
  int* sym = (int*)alloc(sizeof(int) * cB * cK);
  unsigned char* ban0 = (unsigned char*)alloc((size_t)cBK * cV);
  unsigned char* ban1 = (unsigned char*)alloc((size_t)cBK * cV);
  unsigned char* pat = (unsigned char*)alloc(cB * cLM1 * cLM1);
  unsigned char* known = (unsigned char*)alloc(cB * cLM1);

  // Encoder + static init
  k_context<<<256, 128, 0, stream>>>(src, Emb, We, context);
  k_meanctx<<<(cB * cD) / 256, 256, 0, stream>>>(context, meanc);
  k_state0<<<8, 128, 0, stream>>>(meanc, Ws, h);
  k_init_pat<<<(cB * cLM1 * cLM1 + 255) / 256, 256, 0, stream>>>(src, pat);
  k_init_known<<<(cB * cLM1 + 255) / 256, 256, 0, stream>>>(pat, src_len,
                                                            known);
  k_init_state<<<(cBK * cV) / 256, 256, 0, stream>>>(src_len, scores, out0,
                                                     inp, ban0, m0A);

  int* outCur = out0;
  int* outNext = out1;
  unsigned char* banCur = ban0;
  unsigned char* banNext = ban1;
  for (int i = 0; i < cLM1; ++i) {
    k_embed<<<(cBK * cD) / 256, 256, 0, stream>>>(inp, Emb, e);
    k_gru_zr<<<128, 128, 0, stream>>>(e, h, Wz, Uz, Wr, Ur, z, r);
    k_gru_hn<<<64, 128, 0, stream>>>(e, h, r, z, Wn, Un, hn);
    k_attn<<<cBK, 256, 0, stream>>>(hn, context, src, ctxv);
    k_fc<<<64, 128, 0, stream>>>(hn, ctxv, Wc, fh);
    k_mem<<<(cBK * cD) / 256, 256, 0, stream>>>(m0A, e, m0B, mm);
    k_feat<<<64, 128, 0, stream>>>(mm, Wm, fh, feat);
    k_logits<<<800, 160, 0, stream>>>(feat, Wo, logits);
    k_logsoftmax<<<cBK, 256, 0, stream>>>(logits);
    k_topk<<<cB, 256, 0, stream>>>(logits, scores, outCur, known, banCur, src,
                                   i, kidx, sym);
    k_gather_state<<<(cBK * cMEM * cD) / 256, 256, 0, stream>>>(hn, m0B, kidx,
                                                                h, m0A);
    k_gather_ban<<<(cBK * cV) / 256, 256, 0, stream>>>(banCur, kidx, sym,
                                                       banNext, inp);
    k_gather_out<<<(cB * cK * cLM1 + 255) / 256, 256, 0, stream>>>(
        outCur, kidx, sym, pat, i, outNext);
    {
      int* t = outCur;
      outCur = outNext;
      outNext = t;
    }
    {
      unsigned char* t = banCur;
      banCur = banNext;
      banNext = t;
    }
  }
  k_finalize<<<(cB * cK * cLM1 + cB * cK + 255) / 256, 256, 0, stream>>>(
      outCur, scores, (float*)d_out);
}